// MultiModalAttention_32512902430744
// MI455X (gfx1250) — compile-verified
//
#include <hip/hip_runtime.h>

#define H_DIM 128
#define NEG_INF (-3.402823466e38f)

typedef __attribute__((ext_vector_type(2))) float v2f;
typedef __attribute__((ext_vector_type(8))) float v8f;

// D = A(16x4) * B(4x16) + C, fp32 (ISA V_WMMA_F32_16X16X4_F32)
__device__ __forceinline__ v8f wmma_f32_k4(v2f a, v2f b, v8f c) {
  return __builtin_amdgcn_wmma_f32_16x16x4_f32(false, a, false, b, (short)0, c,
                                               false, false);
}

// ---------------------------------------------------------------------------
// Generic WMMA fp32 GEMM:  C[M,Nn] = act(A[M,K] @ W[K,Nn] + bias)
// grid.x = M/16, blockDim = 32*(Nn/16).
// W is staged TRANSPOSED+padded in LDS (WsT[n*(K+2)+k]) so each B fragment is
// one ds_load_b64 at an immediate offset; A tile padded the same way.
// dyn LDS = (Nn*(K+2) + 16*(K+2)) * 4 bytes.
// ---------------------------------------------------------------------------
__global__ __launch_bounds__(256) void gemm_wmma_f32(
    const float* __restrict__ A, const float* __restrict__ W,
    const float* __restrict__ bias, float* __restrict__ C,
    int M, int K, int Nn, int relu_flag) {
  extern __shared__ float smem[];
  const int KA = K + 2;          // padded row stride (even -> 8B alignment)
  float* WsT = smem;             // Nn * KA   (transposed)
  float* As = smem + Nn * KA;    // 16 * KA
  const int tid = threadIdx.x;
  const int nth = blockDim.x;
  const int wave = tid >> 5;
  const int lane = tid & 31;
  const int row0 = blockIdx.x * 16;

  // coalesced global reads; transposed LDS writes (pad avoids bank conflicts)
  for (int i = tid; i < K * Nn; i += nth) {
    const int k = i / Nn, n = i - k * Nn;
    WsT[n * KA + k] = W[i];
  }
  for (int i = tid; i < 16 * K; i += nth) {
    const int r = i / K, k = i - r * K;
    As[r * KA + k] = A[(size_t)(row0 + r) * K + k];
  }
  __syncthreads();

  const int colw = wave * 16;
  const int lhalf = lane >> 4;  // K-half select (ISA A/B 32-bit layout)
  const int l16 = lane & 15;
  const float* ap = As + l16 * KA + lhalf * 2;
  const float* bp = WsT + (colw + l16) * KA + lhalf * 2;
  v8f acc = {};
  for (int kk = 0; kk < K; kk += 4) {
    const v2f a = *(const v2f*)(ap + kk);
    const v2f b = *(const v2f*)(bp + kk);
    acc = wmma_f32_k4(a, b, acc);
  }
  const int n = colw + l16;
  const float bb = bias ? bias[n] : 0.0f;
#pragma unroll
  for (int v = 0; v < 8; ++v) {
    const int m = v + 8 * lhalf;  // C layout: VGPR v -> row v (+8 for hi half)
    float val = acc[v] + bb;
    if (relu_flag) val = fmaxf(val, 0.0f);
    C[(size_t)(row0 + m) * Nn + n] = val;
  }
}

// ---------------------------------------------------------------------------
// Fused edge message + attention scale + scatter:
//   h  = relu(Ya[src] + Yc[dst] + EB[e] + b1) @ W2 + b2
//   agg[dst] += alpha[e] * h          (alpha = expv * inv_sum)
// 16 edges per block, 8 waves (one 16-col slab each), W2 transposed in LDS.
// ---------------------------------------------------------------------------
__global__ __launch_bounds__(256) void propagate_scatter(
    const float* __restrict__ Ya, const float* __restrict__ Yc,
    const float* __restrict__ EB, const float* __restrict__ b1,
    const float* __restrict__ W2, const float* __restrict__ b2,
    const float* __restrict__ expv, const float* __restrict__ inv_sum,
    const int* __restrict__ src, const int* __restrict__ dst,
    float* __restrict__ agg, int E) {
  extern __shared__ float smem[];
  const int KA = H_DIM + 2;
  float* W2s = smem;                        // 128 * KA (transposed, padded)
  float* As = W2s + H_DIM * KA;             // 16 * KA
  float* alphas = As + 16 * KA;             // 16
  int* srcs = (int*)(alphas + 16);          // 16
  int* dsts = srcs + 16;                    // 16

  const int tid = threadIdx.x;
  const int nth = blockDim.x;
  const int wave = tid >> 5;
  const int lane = tid & 31;
  const int e0 = blockIdx.x * 16;
  const float inv = *inv_sum;

  for (int i = tid; i < H_DIM * H_DIM; i += nth) {
    const int k = i >> 7, n = i & (H_DIM - 1);
    W2s[n * KA + k] = W2[i];
  }
  if (tid < 16) {
    const int e = e0 + tid;
    srcs[tid] = src[e];
    dsts[tid] = dst[e];
    alphas[tid] = expv[e] * inv;
  }
  __syncthreads();

  for (int i = tid; i < 16 * H_DIM; i += nth) {
    const int r = i >> 7, k = i & (H_DIM - 1);
    float v = Ya[(size_t)srcs[r] * H_DIM + k] +
              Yc[(size_t)dsts[r] * H_DIM + k] +
              EB[(size_t)(e0 + r) * H_DIM + k] + b1[k];
    As[r * KA + k] = fmaxf(v, 0.0f);
  }
  __syncthreads();

  const int colw = wave * 16;
  const int lhalf = lane >> 4;
  const int l16 = lane & 15;
  const float* ap = As + l16 * KA + lhalf * 2;
  const float* bp = W2s + (colw + l16) * KA + lhalf * 2;
  v8f acc = {};
#pragma unroll
  for (int kk = 0; kk < H_DIM; kk += 4) {
    const v2f a = *(const v2f*)(ap + kk);
    const v2f b = *(const v2f*)(bp + kk);
    acc = wmma_f32_k4(a, b, acc);
  }
  const int n = colw + l16;
  const float bb = b2[n];
#pragma unroll
  for (int v = 0; v < 8; ++v) {
    const int m = v + 8 * lhalf;
    const float val = (acc[v] + bb) * alphas[m];
    atomicAdd(&agg[(size_t)dsts[m] * H_DIM + n], val);
  }
}

// ---------------------------------------------------------------------------
// Attention helper kernels (wave-per-row dot products, wave32 shuffles)
// ---------------------------------------------------------------------------
__global__ void node_att_dots(const float* __restrict__ x,
                              const float* __restrict__ attW,
                              float* __restrict__ la0, float* __restrict__ la1,
                              int N) {
  const int wid = blockIdx.x * (blockDim.x >> 5) + (threadIdx.x >> 5);
  const int lane = threadIdx.x & 31;
  if (wid >= N) return;
  const float* row = x + (size_t)wid * H_DIM;
  float s0 = 0.0f, s1 = 0.0f;
  for (int k = lane; k < H_DIM; k += 32) {
    const float v = row[k];
    s0 += v * attW[k];           // pairs with x_i = x[dst]
    s1 += v * attW[H_DIM + k];   // pairs with x_j = x[src]
  }
  for (int off = 16; off > 0; off >>= 1) {
    s0 += __shfl_down(s0, off, 32);
    s1 += __shfl_down(s1, off, 32);
  }
  if (lane == 0) { la0[wid] = s0; la1[wid] = s1; }
}

__global__ void edge_att_dot(const float* __restrict__ e,
                             const float* __restrict__ attW2,
                             float* __restrict__ elog, int E) {
  const int wid = blockIdx.x * (blockDim.x >> 5) + (threadIdx.x >> 5);
  const int lane = threadIdx.x & 31;
  if (wid >= E) return;
  const float* row = e + (size_t)wid * H_DIM;
  float s = 0.0f;
  for (int k = lane; k < H_DIM; k += 32) s += row[k] * attW2[k];
  for (int off = 16; off > 0; off >>= 1) s += __shfl_down(s, off, 32);
  if (lane == 0) elog[wid] = s;
}

__global__ void edge_logits_k(const float* __restrict__ la0,
                              const float* __restrict__ la1,
                              const float* __restrict__ elog,
                              const int* __restrict__ src,
                              const int* __restrict__ dst,
                              const float* __restrict__ attb,
                              float* __restrict__ logits, int E) {
  const int i = blockIdx.x * blockDim.x + threadIdx.x;
  if (i < E) logits[i] = la0[dst[i]] + la1[src[i]] + elog[i] + *attb;
}

// ---------------------------------------------------------------------------
// Global softmax reductions (two-pass: max, then exp/sum)
// ---------------------------------------------------------------------------
__global__ void reduce_max_partial(const float* __restrict__ v,
                                   float* __restrict__ partial, int n) {
  __shared__ float s[256];
  float m = NEG_INF;
  for (int i = blockIdx.x * blockDim.x + threadIdx.x; i < n;
       i += gridDim.x * blockDim.x)
    m = fmaxf(m, v[i]);
  s[threadIdx.x] = m;
  __syncthreads();
  for (int o = blockDim.x >> 1; o > 0; o >>= 1) {
    if (threadIdx.x < o) s[threadIdx.x] = fmaxf(s[threadIdx.x], s[threadIdx.x + o]);
    __syncthreads();
  }
  if (threadIdx.x == 0) partial[blockIdx.x] = s[0];
}

__global__ void reduce_max_final(const float* __restrict__ partial, int n,
                                 float* __restrict__ out) {
  __shared__ float s[256];
  float m = NEG_INF;
  for (int i = threadIdx.x; i < n; i += blockDim.x) m = fmaxf(m, partial[i]);
  s[threadIdx.x] = m;
  __syncthreads();
  for (int o = blockDim.x >> 1; o > 0; o >>= 1) {
    if (threadIdx.x < o) s[threadIdx.x] = fmaxf(s[threadIdx.x], s[threadIdx.x + o]);
    __syncthreads();
  }
  if (threadIdx.x == 0) out[0] = s[0];
}

__global__ void exp_partial(const float* __restrict__ logits,
                            const float* __restrict__ maxv,
                            float* __restrict__ expv,
                            float* __restrict__ partial, int n) {
  __shared__ float s[256];
  const float m = *maxv;
  float sum = 0.0f;
  for (int i = blockIdx.x * blockDim.x + threadIdx.x; i < n;
       i += gridDim.x * blockDim.x) {
    const float e = __expf(logits[i] - m);
    expv[i] = e;
    sum += e;
  }
  s[threadIdx.x] = sum;
  __syncthreads();
  for (int o = blockDim.x >> 1; o > 0; o >>= 1) {
    if (threadIdx.x < o) s[threadIdx.x] += s[threadIdx.x + o];
    __syncthreads();
  }
  if (threadIdx.x == 0) partial[blockIdx.x] = s[0];
}

__global__ void reduce_sum_final_inv(const float* __restrict__ partial, int n,
                                     float* __restrict__ out_inv) {
  __shared__ float s[256];
  float sum = 0.0f;
  for (int i = threadIdx.x; i < n; i += blockDim.x) sum += partial[i];
  s[threadIdx.x] = sum;
  __syncthreads();
  for (int o = blockDim.x >> 1; o > 0; o >>= 1) {
    if (threadIdx.x < o) s[threadIdx.x] += s[threadIdx.x + o];
    __syncthreads();
  }
  if (threadIdx.x == 0) out_inv[0] = 1.0f / s[0];
}

// ---------------------------------------------------------------------------
// Elementwise / utility kernels
// ---------------------------------------------------------------------------
__global__ void zero_k(float* p, int n) {
  for (int i = blockIdx.x * blockDim.x + threadIdx.x; i < n;
       i += gridDim.x * blockDim.x)
    p[i] = 0.0f;
}
__global__ void copy_k(float* __restrict__ d, const float* __restrict__ s, int n) {
  const int i = blockIdx.x * blockDim.x + threadIdx.x;
  if (i < n) d[i] = s[i];
}
__global__ void add_relu_k(float* __restrict__ x, const float* __restrict__ a, int n) {
  const int i = blockIdx.x * blockDim.x + threadIdx.x;
  if (i < n) x[i] += fmaxf(a[i], 0.0f);   // x = x + relu(prop)
}
__global__ void ts_update_k(float* __restrict__ x, const float* __restrict__ tp,
                            const float* __restrict__ a, float* __restrict__ xsum,
                            int n) {
  const int i = blockIdx.x * blockDim.x + threadIdx.x;
  if (i < n) {
    const float v = fmaxf(tp[i] + a[i], 0.0f);  // x = relu(x@W+b + prop)
    x[i] = v;
    xsum[i] += v;
  }
}
__global__ void scale_k(float* p, float s, int n) {
  const int i = blockIdx.x * blockDim.x + threadIdx.x;
  if (i < n) p[i] *= s;
}
__global__ void pool_k(const float* __restrict__ y, const int* __restrict__ batch,
                       float* __restrict__ pooled, int N) {
  const int i = blockIdx.x * blockDim.x + threadIdx.x;
  if (i < N * H_DIM) {
    const int nrow = i >> 7, c = i & (H_DIM - 1);
    atomicAdd(&pooled[(size_t)batch[nrow] * H_DIM + c], y[i]);
  }
}

// ---------------------------------------------------------------------------
// Host orchestration
// ---------------------------------------------------------------------------
static inline float* carve(char*& p, size_t nfloats) {
  float* r = (float*)p;
  p += nfloats * sizeof(float);
  return r;
}

extern "C" void kernel_launch(void* const* d_in, const int* in_sizes, int n_in,
                              void* d_out, int out_size, void* d_ws,
                              size_t ws_size, hipStream_t stream) {
  (void)in_sizes; (void)out_size; (void)ws_size;
  if (n_in < 32) return;
  const int N = 10000, E = 160000, IN = 64, ED = 16, H = H_DIM, G = 64;

  // Inputs in setup_inputs() dict (insertion) order, params flattened in order.
  const float* x_in  = (const float*)d_in[0];
  const int*   eidx  = (const int*)d_in[1];
  const int*   src   = eidx;        // edge_index[0]
  const int*   dst   = eidx + E;    // edge_index[1]
  const float* eattr = (const float*)d_in[2];
  const int*   batch = (const int*)d_in[3];
  const float* node_W = (const float*)d_in[4];
  const float* node_b = (const float*)d_in[5];
  const float* edge_W1 = (const float*)d_in[6];
  const float* edge_b1 = (const float*)d_in[7];
  const float* edge_W2 = (const float*)d_in[8];
  const float* edge_b2 = (const float*)d_in[9];
  const float* att_W = (const float*)d_in[10];
  const float* att_b = (const float*)d_in[11];
  const float* cW1[3] = {(const float*)d_in[12], (const float*)d_in[16], (const float*)d_in[20]};
  const float* cb1[3] = {(const float*)d_in[13], (const float*)d_in[17], (const float*)d_in[21]};
  const float* cW2[3] = {(const float*)d_in[14], (const float*)d_in[18], (const float*)d_in[22]};
  const float* cb2[3] = {(const float*)d_in[15], (const float*)d_in[19], (const float*)d_in[23]};
  const float* tsW[2] = {(const float*)d_in[24], (const float*)d_in[26]};
  const float* tsb[2] = {(const float*)d_in[25], (const float*)d_in[27]};
  const float* out_W = (const float*)d_in[28];
  const float* out_b = (const float*)d_in[29];
  const float* fin_W = (const float*)d_in[30];
  const float* fin_b = (const float*)d_in[31];

  // Workspace layout
  char* p = (char*)d_ws;
  float* xbuf  = carve(p, (size_t)N * H);
  float* e2    = carve(p, (size_t)E * H);
  float* EB    = carve(p, (size_t)E * H);
  float* Ya    = carve(p, (size_t)N * H);
  float* Yc    = carve(p, (size_t)N * H);
  float* tproj = carve(p, (size_t)N * H);
  float* agg   = carve(p, (size_t)N * H);
  float* xsum  = carve(p, (size_t)N * H);
  float* ybuf  = carve(p, (size_t)N * H);
  float* pooled = carve(p, (size_t)G * H);
  float* e1    = carve(p, (size_t)E * (H / 2));  // only live during edge MLP
  // overlay softmax scratch on e1 (used strictly after e2 is produced)
  float* elog = e1;
  float* logits = e1 + E;
  float* expv = e1 + 2 * (size_t)E;
  float* la0 = e1 + 3 * (size_t)E;
  float* la1 = la0 + N;
  float* partials = la1 + N;   // 256
  float* maxv = partials + 256;
  float* invsum = maxv + 1;

  auto gemm = [&](const float* A, const float* W, const float* bias, float* C,
                  int M, int K, int Nn, int relu) {
    dim3 grid(M / 16), block(32 * (Nn / 16));
    size_t sh = (size_t)((Nn + 16) * (K + 2)) * sizeof(float);
    gemm_wmma_f32<<<grid, block, sh, stream>>>(A, W, bias, C, M, K, Nn, relu);
  };

  const int RB = 240;  // reduction grid
  auto softmax_alpha = [&](const float* xcur) {
    node_att_dots<<<(N + 7) / 8, 256, 0, stream>>>(xcur, att_W, la0, la1, N);
    edge_logits_k<<<(E + 255) / 256, 256, 0, stream>>>(la0, la1, elog, src, dst,
                                                       att_b, logits, E);
    reduce_max_partial<<<RB, 256, 0, stream>>>(logits, partials, E);
    reduce_max_final<<<1, 256, 0, stream>>>(partials, RB, maxv);
    exp_partial<<<RB, 256, 0, stream>>>(logits, maxv, expv, partials, E);
    reduce_sum_final_inv<<<1, 256, 0, stream>>>(partials, RB, invsum);
  };

  auto propagate = [&](const float* xcur, const float* W1, const float* b1,
                       const float* W2, const float* b2, bool computeEB) {
    // per-node halves of the edge MLP first layer (block structure of W1)
    gemm(xcur, W1, nullptr, Ya, N, H, H, 0);                 // x_j block
    gemm(xcur, W1 + 2 * H * H, nullptr, Yc, N, H, H, 0);     // x_i block
    if (computeEB)
      gemm(e2, W1 + H * H, nullptr, EB, E, H, H, 0);         // e block
    softmax_alpha(xcur);
    zero_k<<<512, 256, 0, stream>>>(agg, N * H);
    size_t sh = (size_t)((H + 16) * (H + 2) + 16) * sizeof(float) +
                32 * sizeof(int);
    propagate_scatter<<<E / 16, 256, sh, stream>>>(Ya, Yc, EB, b1, W2, b2, expv,
                                                   invsum, src, dst, agg, E);
  };

  // 1) node / edge embeddings
  gemm(x_in, node_W, node_b, xbuf, N, IN, H, 0);
  gemm(eattr, edge_W1, edge_b1, e1, E, ED, H / 2, 1);
  gemm(e1, edge_W2, edge_b2, e2, E, H / 2, H, 0);
  edge_att_dot<<<(E + 7) / 8, 256, 0, stream>>>(e2, att_W + 2 * H, elog, E);

  // 2) conv layers with residual
  for (int l = 0; l < 3; ++l) {
    propagate(xbuf, cW1[l], cb1[l], cW2[l], cb2[l], true);
    add_relu_k<<<(N * H + 255) / 256, 256, 0, stream>>>(xbuf, agg, N * H);
  }

  // 3) timestep refinement (reuses conv[2]; EB already holds e@W1b of conv[2])
  copy_k<<<(N * H + 255) / 256, 256, 0, stream>>>(xsum, xbuf, N * H);
  for (int t = 0; t < 2; ++t) {
    gemm(xbuf, tsW[t], tsb[t], tproj, N, H, H, 0);
    propagate(xbuf, cW1[2], cb1[2], cW2[2], cb2[2], false);
    ts_update_k<<<(N * H + 255) / 256, 256, 0, stream>>>(xbuf, tproj, agg, xsum,
                                                         N * H);
  }

  // 4) mean over 3 states, output projection, pool, final
  scale_k<<<(N * H + 255) / 256, 256, 0, stream>>>(xsum, 1.0f / 3.0f, N * H);
  gemm(xsum, out_W, out_b, ybuf, N, H, H, 0);
  zero_k<<<32, 256, 0, stream>>>(pooled, G * H);
  pool_k<<<(N * H + 255) / 256, 256, 0, stream>>>(ybuf, batch, pooled, N);
  gemm(pooled, fin_W, fin_b, (float*)d_out, G, H, H, 0);
}